// TypeSpecificAttention_33466385170568
// MI455X (gfx1250) — compile-verified
//
#include <hip/hip_runtime.h>
#include <hip/hip_bf16.h>
#include <math.h>

// Problem constants (match reference)
#define NN   4096   // nodes
#define EE   128    // embed dim
#define HH   8      // heads
#define DD   16     // head dim
#define TT   6      // node types
#define NEDGE 65536 // edges
#define ETT  3      // edge types

typedef __attribute__((ext_vector_type(2))) float v2f;
typedef __attribute__((ext_vector_type(8))) float v8f;

// ---------------------------------------------------------------------------
// Kernel 1: per-type projection  out[n,f] = sum_e x[n,e]*W[t][e][f] + b[t][f]
// One block (128 threads) per node; x row staged in LDS; W row reads are
// contiguous across the 128 threads (weights stay L2-resident: 1.2 MB total).
// ---------------------------------------------------------------------------
__global__ __launch_bounds__(128) void proj_kernel(
    const float* __restrict__ x, const float* __restrict__ W,
    const float* __restrict__ b, const int* __restrict__ types,
    float* __restrict__ out) {
  __shared__ float xs[EE];
  const int n = blockIdx.x;
  const int f = threadIdx.x;
  const int t = types[n];
  xs[f] = x[(size_t)n * EE + f];
  __syncthreads();
  const float* Wt = W + (size_t)t * EE * EE;
  float acc = b[t * EE + f];
#pragma unroll 8
  for (int e = 0; e < EE; ++e) acc = fmaf(xs[e], Wt[(size_t)e * EE + f], acc);
  out[(size_t)n * EE + f] = acc;
}

// ---------------------------------------------------------------------------
// Kernel 2: scores[h,i,j] = (Q[i,h,:] . K[j,h,:]) * 0.25  via WMMA f32 16x16x4
// One wave per 16x16 tile of one head. 8 waves / block. Q/K are L2-resident
// (2 MB each); the 536 MB scores store is the HBM-bound part.
// ---------------------------------------------------------------------------
__global__ __launch_bounds__(256) void scores_kernel(
    const float* __restrict__ Q, const float* __restrict__ K,
    float* __restrict__ attn) {
  const int lane = threadIdx.x & 31;
  const int wave = threadIdx.x >> 5;
  const int w    = blockIdx.x * 8 + wave;        // 524288 waves total
  const int h    = w >> 16;                      // / (256*256)
  const int rem  = w & 0xFFFF;
  const int i0   = (rem >> 8) * 16;
  const int j0   = (rem & 255) * 16;
  const int m    = lane & 15;
  const bool hi  = lane >= 16;

  const float* qrow = Q + (size_t)(i0 + m) * EE + h * DD;
  const float* krow = K + (size_t)(j0 + m) * EE + h * DD;

  v8f c = {};
#pragma unroll
  for (int kb = 0; kb < 4; ++kb) {
    const int k0 = kb * 4 + (hi ? 2 : 0);
    v2f a, bb;
    a.x  = qrow[k0];     a.y  = qrow[k0 + 1];    // A: 16x4 tile of Q
    bb.x = krow[k0];     bb.y = krow[k0 + 1];    // B: 4x16 tile of K^T
    c = __builtin_amdgcn_wmma_f32_16x16x4_f32(false, a, false, bb,
                                              (short)0, c, false, false);
  }
  float* dst = attn + (size_t)h * NN * NN + (size_t)i0 * NN + j0;
#pragma unroll
  for (int r = 0; r < 8; ++r) {
    const int row = hi ? r + 8 : r;
    dst[(size_t)row * NN + m] = c[r] * 0.25f;    // 1/sqrt(16)
  }
}

// ---------------------------------------------------------------------------
// Kernel 3: edge-type bias scatter-add (dup-safe -> f32 global atomics)
// scores[h, tgt, src] += edge_bias[et, h]
// ---------------------------------------------------------------------------
__global__ __launch_bounds__(256) void edge_kernel(
    const int* __restrict__ ei, const int* __restrict__ et,
    const float* __restrict__ ebias, float* __restrict__ attn) {
  const int e   = blockIdx.x * 256 + threadIdx.x;
  const int src = ei[e];
  const int tgt = ei[NEDGE + e];
  const int t   = et[e];
  const size_t base = (size_t)tgt * NN + src;
#pragma unroll
  for (int h = 0; h < HH; ++h)
    atomicAdd(attn + (size_t)h * NN * NN + base, ebias[t * HH + h]);
}

// ---------------------------------------------------------------------------
// Kernel 4 (fused): row softmax in-place + attn-row . V  -> mid[i, h*16+d].
// One 256-thread block per (h,i) row. The row lives in registers (16/thread),
// is normalized, written once to global (required output) and staged in LDS;
// the 16-dim AV output rides for free on this mandatory bandwidth pass,
// eliminating a separate 536 MB re-read of attn.
// ---------------------------------------------------------------------------
__global__ __launch_bounds__(256) void softmax_av_kernel(
    float* __restrict__ attn, const float* __restrict__ V,
    float* __restrict__ mid) {
  const int row = blockIdx.x;            // h*N + i
  const int h   = row >> 12;             // / 4096
  const int i   = row & (NN - 1);
  float* p = attn + (size_t)row * NN;
  const int tid = threadIdx.x;

  __shared__ float rowbuf[NN];           // 16 KB
  __shared__ float redm[8], reds[8];
  __shared__ float part[16][17];         // +1 pad vs bank conflicts

  float v[16];
  float mx = -INFINITY;
#pragma unroll
  for (int k = 0; k < 16; ++k) { v[k] = p[tid + k * 256]; mx = fmaxf(mx, v[k]); }
#pragma unroll
  for (int off = 16; off > 0; off >>= 1) mx = fmaxf(mx, __shfl_xor(mx, off, 32));
  if ((tid & 31) == 0) redm[tid >> 5] = mx;
  __syncthreads();
  float m2 = redm[0];
#pragma unroll
  for (int q = 1; q < 8; ++q) m2 = fmaxf(m2, redm[q]);

  float s = 0.f;
#pragma unroll
  for (int k = 0; k < 16; ++k) { v[k] = __expf(v[k] - m2); s += v[k]; }
#pragma unroll
  for (int off = 16; off > 0; off >>= 1) s += __shfl_xor(s, off, 32);
  if ((tid & 31) == 0) reds[tid >> 5] = s;
  __syncthreads();
  float st = 0.f;
#pragma unroll
  for (int q = 0; q < 8; ++q) st += reds[q];
  const float inv = 1.0f / st;

#pragma unroll
  for (int k = 0; k < 16; ++k) {
    const float t = v[k] * inv;
    p[tid + k * 256]      = t;           // final attn (output #2)
    rowbuf[tid + k * 256] = t;           // stage for AV
  }
  __syncthreads();

  // AV: thread t handles head-dim d = t&15 over j-chunk (t>>4)*256..+255.
  const int d  = tid & 15;
  const int jg = tid >> 4;
  const float* Vh = V + h * DD + d;
  float acc = 0.f;
#pragma unroll 8
  for (int jj = 0; jj < 256; ++jj) {
    const int j = jg * 256 + jj;
    acc = fmaf(rowbuf[j], Vh[(size_t)j * EE], acc);
  }
  part[jg][d] = acc;
  __syncthreads();
  if (tid < 16) {
    float o = 0.f;
#pragma unroll
    for (int g = 0; g < 16; ++g) o += part[g][tid];
    mid[(size_t)i * EE + h * DD + tid] = o;
  }
}

// ---------------------------------------------------------------------------
// Kernel 5: out = mid @ Wo + bo  (4096x128 @ 128x128) via WMMA.
// One wave per 16x16 output tile; K-loop over 128 in steps of 4.
// ---------------------------------------------------------------------------
__global__ __launch_bounds__(256) void out_kernel(
    const float* __restrict__ mid, const float* __restrict__ Wo,
    const float* __restrict__ bo, float* __restrict__ out) {
  const int lane = threadIdx.x & 31;
  const int wave = threadIdx.x >> 5;
  const int w    = blockIdx.x * 8 + wave;        // 2048 waves
  const int i0   = (w >> 3) * 16;
  const int f0   = (w & 7) * 16;
  const int m    = lane & 15;
  const bool hi  = lane >= 16;

  const float* arow = mid + (size_t)(i0 + m) * EE;
  v8f c = {};
#pragma unroll 4
  for (int k = 0; k < EE; k += 4) {
    __builtin_prefetch(arow + k + 32, 0, 0);
    const int ka = k + (hi ? 2 : 0);
    v2f a, bb;
    a.x  = arow[ka];                  a.y  = arow[ka + 1];
    bb.x = Wo[(size_t)ka * EE + f0 + m];
    bb.y = Wo[(size_t)(ka + 1) * EE + f0 + m];
    c = __builtin_amdgcn_wmma_f32_16x16x4_f32(false, a, false, bb,
                                              (short)0, c, false, false);
  }
  const float bias = bo[f0 + m];
#pragma unroll
  for (int r = 0; r < 8; ++r) {
    const int row = hi ? r + 8 : r;
    out[(size_t)(i0 + row) * EE + f0 + m] = c[r] + bias;
  }
}

// ---------------------------------------------------------------------------
extern "C" void kernel_launch(void* const* d_in, const int* in_sizes, int n_in,
                              void* d_out, int out_size, void* d_ws, size_t ws_size,
                              hipStream_t stream) {
  const float* query   = (const float*)d_in[0];
  const float* key_    = (const float*)d_in[1];
  const float* value   = (const float*)d_in[2];
  const float* Wq      = (const float*)d_in[3];
  const float* bq      = (const float*)d_in[4];
  const float* Wk      = (const float*)d_in[5];
  const float* bk      = (const float*)d_in[6];
  const float* Wv      = (const float*)d_in[7];
  const float* bv      = (const float*)d_in[8];
  const float* ebias   = (const float*)d_in[9];
  const float* Wo      = (const float*)d_in[10];
  const float* bo      = (const float*)d_in[11];
  const int*   ntypes  = (const int*)d_in[12];
  const int*   eindex  = (const int*)d_in[13];
  const int*   etypes  = (const int*)d_in[14];

  float* out  = (float*)d_out;                       // [N,E]
  float* attn = out + (size_t)NN * EE;               // [H,N,N] (output #2)

  float* Qp  = (float*)d_ws;                         // [N,E]
  float* Kp  = Qp + (size_t)NN * EE;                 // [N,E]
  float* Vp  = Kp + (size_t)NN * EE;                 // [N,E]
  float* mid = Vp + (size_t)NN * EE;                 // [N,E]

  proj_kernel<<<NN, 128, 0, stream>>>(query, Wq, bq, ntypes, Qp);
  proj_kernel<<<NN, 128, 0, stream>>>(key_,  Wk, bk, ntypes, Kp);
  proj_kernel<<<NN, 128, 0, stream>>>(value, Wv, bv, ntypes, Vp);

  scores_kernel<<<65536, 256, 0, stream>>>(Qp, Kp, attn);        // 524288 waves
  edge_kernel<<<NEDGE / 256, 256, 0, stream>>>(eindex, etypes, ebias, attn);
  softmax_av_kernel<<<HH * NN, 256, 0, stream>>>(attn, Vp, mid); // fused pass
  out_kernel<<<256, 256, 0, stream>>>(mid, Wo, bo, out);         // 2048 waves
}